// DenseBlock_2525440770137
// MI455X (gfx1250) — compile-verified
//
#include <hip/hip_runtime.h>
#include <hip/hip_bf16.h>

// Problem constants (from reference): B=64, D=32, n=128, t=6, leaves=64
#define NB   64
#define DD   32
#define NN   128
#define TT   6
#define NT   (NN*TT)        // 768
#define NCOL (NB*DD)        // 2048
#define LEAF 64
#define JTOT 448            // final concat rows: 64 + 128*3
#define LDX  (JTOT*DD)      // 14336, row stride of output/x buffer per batch

typedef __attribute__((ext_vector_type(2))) float v2f;
typedef __attribute__((ext_vector_type(8))) float v8f;

// ---------------------------------------------------------------- copy x into out rows 0..63
__global__ void k_copy_x(const float* __restrict__ x, float* __restrict__ out) {
    int idx = blockIdx.x * 256 + threadIdx.x;           // 64*64*32 elements
    if (idx >= NB * 64 * DD) return;
    int b   = idx / (64 * DD);
    int rem = idx - b * (64 * DD);
    out[b * LDX + rem] = x[idx];
}

// ---------------------------------------------------------------- sparsemax(fsl) rows of length J
// grid = NT blocks (one per (n,t) row), block = 256.  Rank-based O(J^2) support test.
__global__ void k_fsmax(const float* __restrict__ fsl, float* __restrict__ fs, int J) {
    __shared__ float row[320];
    __shared__ float s_ts[256];
    __shared__ int   s_k[256];
    int nt  = blockIdx.x;
    int tid = threadIdx.x;
    for (int e = tid; e < J; e += 256) row[e] = fsl[nt * J + e];
    __syncthreads();

    float local_ts = 0.0f;
    int   local_k  = 0;
    for (int e = tid; e < J; e += 256) {
        float z = row[e];
        int   c = 1;
        float s = 0.0f;
        for (int j = 0; j < J; ++j) {
            float zj = row[j];
            c += (zj > z) ? 1 : 0;
            s += (zj >= z) ? zj : 0.0f;
        }
        bool supp = ((float)c * z > s - 1.0f);
        local_k  += supp ? 1 : 0;
        local_ts += supp ? z : 0.0f;
    }
    s_ts[tid] = local_ts;
    s_k[tid]  = local_k;
    __syncthreads();
    for (int off = 128; off > 0; off >>= 1) {
        if (tid < off) { s_ts[tid] += s_ts[tid + off]; s_k[tid] += s_k[tid + off]; }
        __syncthreads();
    }
    float tau = (s_ts[0] - 1.0f) / (float)s_k[0];
    for (int e = tid; e < J; e += 256) fs[nt * J + e] = fmaxf(row[e] - tau, 0.0f);
}

// ---------------------------------------------------------------- essum[nt][d] = sum_j sparsemax(esl[nt,j,:])[d]
// Row-per-lane: each wave processes a 32-row tile; each lane sorts one row in
// registers (unrolled bitonic, 240 comparators) -> 32 sparsemaxes per ~800 instrs.
// grid = NT blocks, block = 256 = 8 waves; wave w handles tiles w, w+8, ...
#define TPAD 33   // LDS tile row stride (gcd(33,64)=1 -> conflict-free both ways)
__global__ void k_essum(const float* __restrict__ esl, float* __restrict__ essum, int J) {
    __shared__ float tile[8][32 * TPAD];
    __shared__ float accs[8][32];
    int nt   = blockIdx.x;
    int lane = threadIdx.x & 31;
    int wid  = threadIdx.x >> 5;
    float* my = tile[wid];
    float acc = 0.0f;
    int ntiles = J >> 5;
    for (int tb = wid; tb < ntiles; tb += 8) {
        size_t rowbase = (size_t)nt * J + tb * 32;
        // coalesced load of 32x32 tile into padded LDS (wave-private slice)
        #pragma unroll
        for (int rr = 0; rr < 32; ++rr) {
            my[rr * TPAD + lane] = esl[(rowbase + rr) * DD + lane];
        }
        // each lane pulls its own row (conflict-free: stride 33 mod 64 banks)
        float r[32];
        #pragma unroll
        for (int k = 0; k < 32; ++k) r[k] = my[lane * TPAD + k];
        // bitonic sort, descending, fully unrolled in registers
        #pragma unroll
        for (int k = 2; k <= 32; k <<= 1) {
            #pragma unroll
            for (int j = k >> 1; j > 0; j >>= 1) {
                #pragma unroll
                for (int i = 0; i < 32; ++i) {
                    int ij = i ^ j;
                    if (ij > i) {
                        float a  = r[i], b2 = r[ij];
                        float lo = fminf(a, b2), hi = fmaxf(a, b2);
                        if ((i & k) == 0) { r[i] = hi; r[ij] = lo; }
                        else              { r[i] = lo; r[ij] = hi; }
                    }
                }
            }
        }
        // sparsemax support scan (indicators are prefix-true)
        float cum = 0.0f, tausum = 1.0f;
        int   ksup = 1;
        #pragma unroll
        for (int i = 0; i < 32; ++i) {
            cum += r[i];
            bool s = ((float)(i + 1) * r[i] > cum - 1.0f);
            if (s) { ksup = i + 1; tausum = cum; }
        }
        float tau = (tausum - 1.0f) / (float)ksup;   // tau for row (tb*32 + lane)
        // accumulate relu(z - tau_row) in the coalesced orientation
        #pragma unroll
        for (int j2 = 0; j2 < 32; ++j2) {
            float z = my[j2 * TPAD + lane];
            float t = __shfl(tau, j2, 32);
            acc += fmaxf(z - t, 0.0f);
        }
    }
    accs[wid][lane] = acc;
    __syncthreads();
    if (threadIdx.x < 32) {
        float s = 0.0f;
        #pragma unroll
        for (int w = 0; w < 8; ++w) s += accs[w][threadIdx.x];
        essum[nt * DD + threadIdx.x] = s;
    }
}

// ---------------------------------------------------------------- WMMA GEMM + epilogue
// sel[nt,col] = sum_k fs[nt,k] * x[b,k,d]  (col = b*32+d), then
// tl = (sel*essum[nt,d] - thr[nt]) * exp(-lt[nt])
// M=768, N=2048, K=J. One wave per 16x16 tile; v_wmma_f32_16x16x4_f32 over K.
__global__ void k_gemm_tl(const float* __restrict__ x, const float* __restrict__ fs,
                          const float* __restrict__ essum, const float* __restrict__ thr,
                          const float* __restrict__ lt, float* __restrict__ tl, int K) {
    int lane  = threadIdx.x & 31;
    int wid   = threadIdx.x >> 5;
    int tile  = blockIdx.x * 8 + wid;
    int mt    = tile >> 7;      // 0..47
    int ntile = tile & 127;     // 0..127
    int m0    = mt * 16;
    int col   = ntile * 16 + (lane & 15);
    int bidx  = col >> 5;
    int d     = col & 31;
    int ka_off = (lane >> 4) << 1;   // lanes 0-15 -> K {k,k+1}; lanes 16-31 -> {k+2,k+3}

    const float* arow = fs + (size_t)(m0 + (lane & 15)) * K;
    const float* bcol = x + (size_t)bidx * LDX + d;

    v8f acc = {};
    for (int k0 = 0; k0 < K; k0 += 4) {
        int ka = k0 + ka_off;
        v2f a;
        a.x = arow[ka];
        a.y = arow[ka + 1];
        v2f b;
        b.x = bcol[ka * DD];
        b.y = bcol[(ka + 1) * DD];
        acc = __builtin_amdgcn_wmma_f32_16x16x4_f32(false, a, false, b,
                                                    (short)0, acc, false, false);
    }
    // C/D layout: VGPR v -> row m0 + v (lanes 0-15) or m0 + v + 8 (lanes 16-31), N = col
    int mplus = (lane >> 4) << 3;
    #pragma unroll
    for (int v = 0; v < 8; ++v) {
        int nt   = m0 + v + mplus;
        float sel = acc[v] * essum[nt * DD + d];
        float tv  = (sel - thr[nt]) * __expf(-lt[nt]);
        tl[(size_t)nt * NCOL + col] = tv;
    }
}

// ---------------------------------------------------------------- leaf product tree + response dot
// block = (nn, 8-batch group); resp[nn] staged once in LDS (padded stride 65).
// wave w -> b = bg*8 + w; lane = d.  grid = 128*8 = 1024 blocks of 256.
__global__ void k_tree(const float* __restrict__ tl, const float* __restrict__ resp,
                       float* __restrict__ out, int rowOff) {
    __shared__ float rsp[DD * 65];
    int nn  = blockIdx.x >> 3;
    int bg  = blockIdx.x & 7;
    int tid = threadIdx.x;
    for (int idx = tid; idx < DD * LEAF; idx += 256) {
        int dd = idx >> 6, cc = idx & 63;
        rsp[dd * 65 + cc] = resp[(size_t)nn * (DD * LEAF) + idx];
    }
    __syncthreads();

    int lane = tid & 31;
    int wid  = tid >> 5;
    int b    = bg * 8 + wid;

    float p[TT], q[TT];
    #pragma unroll
    for (int t = 0; t < TT; ++t) {
        float v = tl[(size_t)(nn * TT + t) * NCOL + b * DD + lane];
        p[t] = fminf(fmaxf(0.5f * v + 0.5f, 0.0f), 1.0f);   // code bit = 0
        q[t] = fminf(fmaxf(0.5f - 0.5f * v, 0.0f), 1.0f);   // code bit = 1
    }
    float r[LEAF];
    r[0] = 1.0f;
    #pragma unroll
    for (int t = 0; t < TT; ++t) {
        int sz = 1 << t;
        #pragma unroll
        for (int c = 0; c < 32; ++c) {   // upper bound; guarded by sz
            int cc = sz - 1 - c;
            if (cc >= 0) {
                r[cc + sz] = r[cc] * q[t];
                r[cc]      = r[cc] * p[t];
            }
        }
    }
    float s = 0.0f;
    #pragma unroll
    for (int c = 0; c < LEAF; ++c) s += r[c] * rsp[lane * 65 + c];
    out[(size_t)b * LDX + (rowOff + nn) * DD + lane] = s;
}

// ----------------------------------------------------------------
extern "C" void kernel_launch(void* const* d_in, const int* in_sizes, int n_in,
                              void* d_out, int out_size, void* d_ws, size_t ws_size,
                              hipStream_t stream) {
    const float* x = (const float*)d_in[0];
    float* out = (float*)d_out;

    float* fs_ws    = (float*)d_ws;                 // 768*320
    float* essum_ws = fs_ws + NT * 320;             // 768*32
    float* tl_ws    = essum_ws + NT * DD;           // 768*2048

    k_copy_x<<<(NB * 64 * DD + 255) / 256, 256, 0, stream>>>(x, out);

    const int Js[3]     = {64, 192, 320};
    const int rowOff[3] = {64, 192, 320};
    for (int i = 0; i < 3; ++i) {
        const float* fsl  = (const float*)d_in[1 + 5 * i];
        const float* esl  = (const float*)d_in[2 + 5 * i];
        const float* thr  = (const float*)d_in[3 + 5 * i];
        const float* lt   = (const float*)d_in[4 + 5 * i];
        const float* resp = (const float*)d_in[5 + 5 * i];
        int J = Js[i];

        k_fsmax <<<NT, 256, 0, stream>>>(fsl, fs_ws, J);
        k_essum <<<NT, 256, 0, stream>>>(esl, essum_ws, J);
        k_gemm_tl<<<(48 * 128) / 8, 256, 0, stream>>>(out, fs_ws, essum_ws, thr, lt, tl_ws, J);
        k_tree  <<<NN * 8, 256, 0, stream>>>(tl_ws, resp, out, rowOff[i]);
    }
}